// ResidualVectorQuantizer_43679817401105
// MI455X (gfx1250) — compile-verified
//
#include <hip/hip_runtime.h>
#include <hip/hip_bf16.h>
#include <cstdint>
#include <cstddef>

// ---------------------------------------------------------------------------
// Residual Vector Quantizer for MI455X (gfx1250, wave32, WMMA).
//  - V_WMMA_F32_16X16X4_F32 for the (N x 128) x (128 x 1024) distance GEMM
//  - code-norm table precomputed once (lives in L2)
//  - double-buffered code tiles; async global->LDS copies (ASYNCcnt-tracked)
//  - z_q output via telescoping: sum(zq_s) == z_e - final_residual
//    (removes 64 VGPRs of accumulators -> lets the scheduler pipeline DS loads)
// ---------------------------------------------------------------------------

typedef __attribute__((ext_vector_type(2))) float v2f;
typedef __attribute__((ext_vector_type(8))) float v8f;
typedef int v4i __attribute__((ext_vector_type(4)));

#define B_      16
#define D_      128
#define T_      2048
#define N_      (B_ * T_)        // 32768 vectors
#define NQ      8
#define KCODES  1024
#define BM      128              // rows per block
#define CT      128              // codes per LDS tile
#define NT      (KCODES / CT)    // 8 tiles per stage
#define SD      132              // padded LDS row stride (floats) -> conflict-free
#define NTHREADS 256
#define NWAVES   8
#define COMMIT_W_F 1.0f

// flat output offsets (floats)
#define OUT_IDX     ((size_t)B_ * D_ * T_)            // 4194304
#define OUT_SCALAR  (OUT_IDX + (size_t)B_ * NQ * T_)  // 4456448

// workspace layout (bytes): commitPart | counts | e2all
#define WS_COMMIT_FLOATS (NQ * 256)
#define WS_COUNT_WORDS   (NQ * KCODES)
#define WS_E2_FLOATS     (NQ * KCODES)
#define WS_COUNT_OFF     (WS_COMMIT_FLOATS * 4)
#define WS_E2_OFF        (WS_COUNT_OFF + WS_COUNT_WORDS * 4)

// LDS: resid | codes(x2) | bidx | redBuf
#define SMEM_FLOATS (BM * SD + 2 * CT * SD)
#define SMEM_BYTES  ((SMEM_FLOATS) * 4 + BM * 4 + NWAVES * 4)

// ---- CDNA5 async global->LDS path (probe-guarded; sync fallback otherwise)
#if defined(__has_builtin)
#  if __has_builtin(__builtin_amdgcn_global_load_async_to_lds_b128) && \
      __has_builtin(__builtin_amdgcn_s_wait_asynccnt)
#    define HAVE_ASYNC_LDS 1
#  endif
#endif
#ifndef HAVE_ASYNC_LDS
#  define HAVE_ASYNC_LDS 0
#endif

#if HAVE_ASYNC_LDS
typedef __attribute__((address_space(1))) v4i as1_v4i;   // global 128-bit chunk
typedef __attribute__((address_space(3))) v4i as3_v4i;   // LDS    128-bit chunk
#endif

__global__ void rvq_init(float* __restrict__ commitPart, unsigned* __restrict__ counts) {
    int i = blockIdx.x * blockDim.x + threadIdx.x;
    if (i < WS_COMMIT_FLOATS) commitPart[i] = 0.0f;
    if (i < WS_COUNT_WORDS)   counts[i] = 0u;
}

// one wave per code row: e2all[row] = sum_d emb[row][d]^2
__global__ void rvq_e2(const float* __restrict__ emb, float* __restrict__ e2all) {
    const int wave = threadIdx.x >> 5;
    const int lane = threadIdx.x & 31;
    const int row  = blockIdx.x * NWAVES + wave;     // NQ*KCODES rows total
    const float4 v = ((const float4*)(emb + (size_t)row * D_))[lane];
    float s = v.x * v.x + v.y * v.y + v.z * v.z + v.w * v.w;
#pragma unroll
    for (int m = 1; m < 32; m <<= 1) s += __shfl_xor(s, m, 32);
    if (lane == 0) e2all[row] = s;
}

// copy one 128x128 f32 code tile into LDS (async when available)
static __device__ __forceinline__
void issue_tile(const float* __restrict__ src, float* __restrict__ dstLds, int tid) {
#if HAVE_ASYNC_LDS
    for (int it = 0; it < (CT * (D_ / 4)) / NTHREADS; ++it) {   // 16 per thread
        int lin = tid + it * NTHREADS;
        int c   = lin >> 5;
        int d4  = lin & 31;
        __builtin_amdgcn_global_load_async_to_lds_b128(
            (as1_v4i*)(src + (size_t)c * D_ + d4 * 4),
            (as3_v4i*)(dstLds + c * SD + d4 * 4),
            0, 0);
    }
#else
    for (int lin = tid; lin < CT * (D_ / 4); lin += NTHREADS) {
        int c  = lin >> 5;
        int d4 = lin & 31;
        const float4 v = ((const float4*)src)[c * 32 + d4];
        float* cp = dstLds + c * SD + d4 * 4;
        cp[0] = v.x; cp[1] = v.y; cp[2] = v.z; cp[3] = v.w;
    }
    __builtin_prefetch(src + (size_t)CT * D_ + tid * 64, 0, 1);
#endif
}

static __device__ __forceinline__ void wait_tile(bool more_inflight) {
#if HAVE_ASYNC_LDS
    if (more_inflight) __builtin_amdgcn_s_wait_asynccnt(16);
    else               __builtin_amdgcn_s_wait_asynccnt(0);
#else
    (void)more_inflight;
#endif
}

// 16 rows x 128 codes dot-product tile; `codes` must be a compile-time offset
// from the LDS base at every call site (keeps DS immediate-offset addressing).
static __device__ __forceinline__
void gemm_tile(const float* __restrict__ aBase, const float* __restrict__ codes,
               int l16, int half, v8f acc[8]) {
#pragma unroll 4
    for (int kk = 0; kk < D_; kk += 4) {
        const int ka = kk + 2 * half;
        const v2f a = *(const v2f*)(aBase + ka);
#pragma unroll
        for (int j = 0; j < 8; ++j) {
            const v2f b = *(const v2f*)(codes + (j * 16 + l16) * SD + ka);
            acc[j] = __builtin_amdgcn_wmma_f32_16x16x4_f32(
                false, a, false, b, (short)0, acc[j], false, false);
        }
    }
}

__global__ __launch_bounds__(NTHREADS)
void rvq_main(const float* __restrict__ zE, const float* __restrict__ emb,
              const float* __restrict__ e2all,
              float* __restrict__ out, float* __restrict__ commitPart,
              unsigned* __restrict__ counts) {
    extern __shared__ float smem[];
    float* resid     = smem;                          // BM*SD
    float* codesBase = resid + BM * SD;               // 2 * CT*SD (double buffer)
    int*   bidx      = (int*)(codesBase + 2 * CT * SD);
    float* redBuf    = (float*)(bidx + BM);           // NWAVES

    const int tid  = threadIdx.x;
    const int wave = tid >> 5;
    const int lane = tid & 31;
    const int half = lane >> 4;
    const int l16  = lane & 15;

    const int n0    = blockIdx.x * BM;                // 128-row tile, within one b
    const int bIdx  = n0 / T_;
    const int tbase = n0 % T_;
    const float* zeB = zE + (size_t)bIdx * D_ * T_ + tbase;

    // ---- load residual tile: resid[r][d] = z_e[b][d][tbase+r] (coalesced in t)
    for (int lin = tid; lin < BM * D_; lin += NTHREADS) {
        int d = lin >> 7;
        int r = lin & 127;
        resid[r * SD + d] = zeB[(size_t)d * T_ + r];
    }

    const v8f zero8 = {0.f, 0.f, 0.f, 0.f, 0.f, 0.f, 0.f, 0.f};
    const int rb = wave * 16;                         // wave's 16 rows

    for (int s = 0; s < NQ; ++s) {
        const float* embS = emb + (size_t)s * KCODES * D_;
        const float* e2S  = e2all + (size_t)s * KCODES;

        float bv[8];
        int   bi[8];
#pragma unroll
        for (int r = 0; r < 8; ++r) { bv[r] = 3.4e38f; bi[r] = 0; }

        // prologue: start streaming tile 0 into buffer 0
        issue_tile(embS, codesBase, tid);

        for (int ct = 0; ct < NT; ++ct) {
            // overlap: start next tile while current GEMM runs
            if (ct + 1 < NT)
                issue_tile(embS + (size_t)(ct + 1) * CT * D_,
                           codesBase + ((ct + 1) & 1) * (CT * SD), tid);
            wait_tile(ct + 1 < NT);   // tile ct landed (async loads retire in order)
            __syncthreads();

            // ---- GEMM via v_wmma_f32_16x16x4_f32 (constant LDS base per site)
            v8f acc[8];
#pragma unroll
            for (int j = 0; j < 8; ++j) acc[j] = zero8;

            const float* aBase = resid + (rb + l16) * SD;
            if ((ct & 1) == 0) gemm_tile(aBase, codesBase, l16, half, acc);
            else               gemm_tile(aBase, codesBase + CT * SD, l16, half, acc);

            // ---- fold tile into running argmin: dist ~ e2[c] - 2*dot
#pragma unroll
            for (int j = 0; j < 8; ++j) {
                const int   code = ct * CT + j * 16 + l16;
                const float e    = e2S[code];
#pragma unroll
                for (int r = 0; r < 8; ++r) {
                    float val = e - 2.0f * acc[j][r];
                    if (val < bv[r] || (val == bv[r] && code < bi[r])) {
                        bv[r] = val; bi[r] = code;
                    }
                }
            }
            __syncthreads();   // all waves done with this buffer before re-fill
        } // code tiles

        // ---- cross-lane argmin inside each 16-lane half (first-index tiebreak)
#pragma unroll
        for (int r = 0; r < 8; ++r) {
            float v = bv[r];
            int  ix = bi[r];
#pragma unroll
            for (int m = 1; m <= 8; m <<= 1) {
                float ov = __shfl_xor(v, m, 32);
                int   oi = __shfl_xor(ix, m, 32);
                if (ov < v || (ov == v && oi < ix)) { v = ov; ix = oi; }
            }
            if (l16 == 0) bidx[rb + half * 8 + r] = ix;
        }
        __syncthreads();

        // ---- gather z_q, update residual in LDS, accumulate commit
        float commitLocal = 0.f;
#pragma unroll
        for (int i = 0; i < 16; ++i) {
            int lin = tid + i * NTHREADS;
            int row = lin >> 5;
            int d4  = lin & 31;
            int code = bidx[row];
            const float4 q = ((const float4*)(embS + (size_t)code * D_))[d4];
            float* rp = resid + row * SD + d4 * 4;
            float r0 = rp[0] - q.x; rp[0] = r0;
            float r1 = rp[1] - q.y; rp[1] = r1;
            float r2 = rp[2] - q.z; rp[2] = r2;
            float r3 = rp[3] - q.w; rp[3] = r3;
            commitLocal += r0 * r0 + r1 * r1 + r2 * r2 + r3 * r3;
        }

        // ---- per-row outputs: indices + code-usage counts
        if (tid < BM) {
            int code = bidx[tid];
            out[OUT_IDX + (size_t)bIdx * NQ * T_ + (size_t)s * T_ + (tbase + tid)] = (float)code;
            atomicAdd(&counts[s * KCODES + code], 1u);
        }

        // ---- deterministic commit partial: one slot per (stage, block)
#pragma unroll
        for (int m = 1; m < 32; m <<= 1) commitLocal += __shfl_xor(commitLocal, m, 32);
        if (lane == 0) redBuf[wave] = commitLocal;
        __syncthreads();
        if (tid == 0) {
            float sum = 0.f;
            for (int w = 0; w < NWAVES; ++w) sum += redBuf[w];
            commitPart[s * gridDim.x + blockIdx.x] = sum;
        }
        __syncthreads();
    } // stages

    // ---- z_q output by telescoping: sum(zq_s) = z_e - final_residual
    // (z_e re-read from L2; both global streams coalesced along t)
    for (int lin = tid; lin < BM * D_; lin += NTHREADS) {
        int d = lin >> 7;
        int r = lin & 127;
        out[(size_t)bIdx * D_ * T_ + (size_t)d * T_ + tbase + r] =
            zeB[(size_t)d * T_ + r] - resid[r * SD + d];
    }
}

__global__ void rvq_finalize(const float* __restrict__ commitPart,
                             const unsigned* __restrict__ counts,
                             float* __restrict__ out) {
    __shared__ float fsum[256];
    __shared__ int   isum[256];
    int tid = threadIdx.x;
    float cs = 0.f;
    for (int i = tid; i < WS_COMMIT_FLOATS; i += 256) cs += commitPart[i];
    int nz = 0;
    for (int i = tid; i < WS_COUNT_WORDS; i += 256) nz += (counts[i] != 0u) ? 1 : 0;
    fsum[tid] = cs; isum[tid] = nz;
    __syncthreads();
    for (int st = 128; st > 0; st >>= 1) {
        if (tid < st) { fsum[tid] += fsum[tid + st]; isum[tid] += isum[tid + st]; }
        __syncthreads();
    }
    if (tid == 0) {
        out[OUT_SCALAR + 0] = COMMIT_W_F * fsum[0] / ((float)NQ * (float)N_ * (float)D_);
        out[OUT_SCALAR + 1] = (float)isum[0] / ((float)NQ * (float)KCODES);
    }
}

extern "C" void kernel_launch(void* const* d_in, const int* in_sizes, int n_in,
                              void* d_out, int out_size, void* d_ws, size_t ws_size,
                              hipStream_t stream) {
    const float* zE  = (const float*)d_in[0];   // (16, 128, 2048) f32
    const float* emb = (const float*)d_in[1];   // (8, 1024, 128) f32
    float* out = (float*)d_out;

    float*    commitPart = (float*)d_ws;
    unsigned* counts     = (unsigned*)((char*)d_ws + WS_COUNT_OFF);
    float*    e2all      = (float*)((char*)d_ws + WS_E2_OFF);

    (void)hipFuncSetAttribute((const void*)rvq_main,
                              hipFuncAttributeMaxDynamicSharedMemorySize, SMEM_BYTES);

    rvq_init<<<(WS_COUNT_WORDS + 255) / 256, 256, 0, stream>>>(commitPart, counts);
    rvq_e2<<<(NQ * KCODES) / NWAVES, 256, 0, stream>>>(emb, e2all);
    rvq_main<<<N_ / BM, NTHREADS, SMEM_BYTES, stream>>>(zE, emb, e2all, out, commitPart, counts);
    rvq_finalize<<<1, 256, 0, stream>>>(commitPart, counts, out);
}